// GRNN_19550691131429
// MI455X (gfx1250) — compile-verified
//
#include <hip/hip_runtime.h>
#include <hip/hip_bf16.h>
#include <stdint.h>

// Quantum-trajectory Kalman/Riccati scan.
// B=512 trajectories, T=4096 sequential steps. One trajectory per lane.
// dy chunks are staged into LDS with CDNA5 async global->LDS DMA
// (global_load_async_to_lds_b128, ASYNCcnt), double-buffered per wave.

#define BATCH   512
#define TSTEPS  4096
#define BLOCK   32            // one wave per workgroup -> barrier-free staging
#define CH      64            // time-steps per staged chunk
#define NCHUNK  (TSTEPS / CH) // 64
#define JPC     (CH / 2)      // 32 x b128 async loads per chunk (2 steps / 16B)

// One SDE step. Only dy[...,0] is live (C has a single nonzero entry),
// dy_hat[...,1] == 0 always.
#define STEP(dy0, yh)                                                        \
  do {                                                                       \
    yh = (Cc * x0) * DTf;                                                    \
    float m00 = -(0.5f * G) - C2 * vx;                                       \
    float m10 = -w - C2 * cxp;                                               \
    float dx0 = (m00 * x0 + w * x1) * DTf + (Cc * vx) * (dy0);               \
    float dx1 = (m10 * x0 - (0.5f * G) * x1) * DTf + (Cc * cxp) * (dy0);     \
    float nvx = vx + DTf * (Dc - G * vx + 2.0f * (w * cxp) - C2 * (vx * vx));\
    float nvp = vp + DTf * (Dc - G * vp - 2.0f * (w * cxp) - C2 * (cxp*cxp));\
    float ncx = cxp + DTf * (w * (vp - vx) - G * cxp - C2 * (vx * cxp));     \
    x0 += dx0; x1 += dx1; vx = nvx; vp = nvp; cxp = ncx; t += DTf;           \
  } while (0)

__global__ __launch_bounds__(BLOCK) void kf_scan_kernel(
    const float* __restrict__ dy,       // (B, T, 2)
    const float* __restrict__ state0,   // (B, 6): x0,x1,vx,vp,cxp,t
    const float* __restrict__ omega_p,  // (1,)
    float* __restrict__ out)            // (B*6) final_state ++ (B*T*2) dy_hat
{
  // Double buffer: [2][JPC*BLOCK] float4 = 2 * 32*32*16B = 32 KB LDS.
  // float4 = {dy0(t), dy1(t), dy0(t+1), dy1(t+1)} for this lane.
  __shared__ float4 lbuf[2][JPC * BLOCK];

  const int lane = (int)threadIdx.x;
  const int b    = (int)blockIdx.x * BLOCK + lane;

  const float DTf = 0.001f;
  const float G   = 0.3f;
  const float Cc  = 1.6970562748477140585f; // sqrt(4*ETA*KAPPA) = sqrt(2.88)
  const float C2  = 2.88f;                  // Cc*Cc
  const float Dc  = 1.65f;                  // GAMMA*(NBAR+0.5)+KAPPA
  const float w   = omega_p[0];

  float x0  = state0[b * 6 + 0];
  float x1  = state0[b * 6 + 1];
  float vx  = state0[b * 6 + 2];
  float vp  = state0[b * 6 + 3];
  float cxp = state0[b * 6 + 4];
  float t   = state0[b * 6 + 5];

  // Per-lane global base of this trajectory's dy stream (contiguous 32KB).
  const uint64_t gbase = (uint64_t)(uintptr_t)(dy + (size_t)b * (TSTEPS * 2));
  float* const dyh = out + BATCH * 6 + (size_t)b * (TSTEPS * 2);

  // LDS byte addresses (flat shared pointer low 32 bits == LDS offset).
  const uint32_t lb0 = (uint32_t)(uintptr_t)&lbuf[0][lane];
  const uint32_t lb1 = (uint32_t)(uintptr_t)&lbuf[1][lane];

  // Issue one chunk: JPC async b128 gathers (each lane: 16B of its stream).
  auto issue_chunk = [&](int c, uint32_t lb) {
    const uint64_t ga = gbase + (uint64_t)c * (CH * 8);
#pragma unroll
    for (int j = 0; j < JPC; ++j) {
      asm volatile("global_load_async_to_lds_b128 %0, %1, off"
                   :: "v"(lb + (uint32_t)(j * (BLOCK * 16))),
                      "v"(ga + (uint64_t)(j * 16))
                   : "memory");
    }
  };

  issue_chunk(0, lb0);

  for (int c = 0; c < NCHUNK; ++c) {
    const int buf = c & 1;
    if (c + 1 < NCHUNK) {
      // Prefetch next chunk into the other buffer, then wait until only
      // those JPC(=32) loads remain outstanding (in-order completion =>
      // chunk c is fully landed in LDS).
      issue_chunk(c + 1, buf ? lb0 : lb1);
      asm volatile("s_wait_asynccnt 32" ::: "memory");
    } else {
      asm volatile("s_wait_asynccnt 0" ::: "memory");
    }

    const float4* __restrict__ mybuf = &lbuf[buf][lane];
    float* __restrict__ outp = dyh + (size_t)(c * CH) * 2;

#pragma unroll 4
    for (int j = 0; j < JPC; ++j) {
      const float4 v = mybuf[(size_t)j * BLOCK];  // ds_load_b128
      float yh0, yh1;
      STEP(v.x, yh0);   // step 2j   (uses dy0 only)
      STEP(v.z, yh1);   // step 2j+1
      float4 o;
      o.x = yh0; o.y = 0.0f; o.z = yh1; o.w = 0.0f;
      *(float4*)(outp + (size_t)j * 4) = o;       // global_store_b128, 16B aligned
    }
  }

  // Final state: [x0, x1, vx, vp, cxp, t]
  out[b * 6 + 0] = x0;
  out[b * 6 + 1] = x1;
  out[b * 6 + 2] = vx;
  out[b * 6 + 3] = vp;
  out[b * 6 + 4] = cxp;
  out[b * 6 + 5] = t;
}

extern "C" void kernel_launch(void* const* d_in, const int* in_sizes, int n_in,
                              void* d_out, int out_size, void* d_ws, size_t ws_size,
                              hipStream_t stream) {
  (void)in_sizes; (void)n_in; (void)d_ws; (void)ws_size; (void)out_size;
  const float* dy      = (const float*)d_in[0];  // (512, 4096, 2) f32
  const float* state0  = (const float*)d_in[1];  // (512, 6) f32
  const float* omega_p = (const float*)d_in[2];  // (1,) f32
  float* out = (float*)d_out;                    // 512*6 ++ 512*4096*2 f32

  kf_scan_kernel<<<BATCH / BLOCK, BLOCK, 0, stream>>>(dy, state0, omega_p, out);
}